// HG2VecModel_29738353557505
// MI455X (gfx1250) — compile-verified
//
#include <hip/hip_runtime.h>
#include <hip/hip_bf16.h>

typedef float v2f __attribute__((ext_vector_type(2)));
typedef float v8f __attribute__((ext_vector_type(8)));

#define DIM   300
#define NB    64
#define NL    256
#define NC    10
#define NI    6
#define TILES (NB * NL)          /* 16384 (b,l) sites, T==1 */
#define WPB   8                  /* waves (tiles) per block   */
#define NBLK  (TILES / WPB)      /* 2048 blocks               */

// One wave32 per (b,l) tile.  A(16x300) x B(300x16) -> D(16x16) via 75
// v_wmma_f32_16x16x4_f32 steps.  D(0, 0..9) = score, D(1+c, 10+i) = info_score.
__global__ __launch_bounds__(256) void hg2vec_wmma_kernel(
    const int*   __restrict__ pos_u,   // [TILES]
    const int*   __restrict__ pos_v,   // [TILES*NC]
    const int*   __restrict__ info_v,  // [TILES*NI]
    const float* __restrict__ W_in,    // [V, DIM]
    const float* __restrict__ W_out,   // [V, DIM]
    const float* __restrict__ cmask,   // [NC]
    const float* __restrict__ smask,   // [NI]  (+-1 sign mask)
    const float* __restrict__ scmask,  // [NI]  (score mask)
    float*       __restrict__ partials)// [NBLK]
{
    const int lane = threadIdx.x & 31;
    const int wave = threadIdx.x >> 5;
    const int tile = blockIdx.x * WPB + wave;
    const int rid  = lane & 15;          // A row == B col handled by this lane
    const int koff = (lane >> 4) << 1;   // 0 for lanes 0-15, 2 for lanes 16-31

    // ---- A-matrix source row (from W_out) ----
    int a_idx;
    if (rid == 0)        a_idx = pos_u[tile];
    else if (rid <= NC)  a_idx = pos_v[tile * NC + (rid - 1)];
    else                 a_idx = 0;      // vocab row 0 is all zeros -> padding
    const float* arow = W_out + (size_t)a_idx * DIM;

    // ---- B-matrix source row (from W_in) ----
    int b_idx;
    if (rid < NC) b_idx = pos_v[tile * NC + rid];
    else          b_idx = info_v[tile * NI + (rid - NC)];
    const float* brow = W_in + (size_t)b_idx * DIM;

    // Warm the gathered rows (global_prefetch_b8)
    __builtin_prefetch(arow, 0, 3);
    __builtin_prefetch(brow, 0, 3);

    v8f acc = {};
#pragma unroll 5
    for (int k = 0; k < DIM; k += 4) {
        // 16x4 A frag & 4x16 B frag: lane holds elements [k+koff, k+koff+1]
        const v2f a = *(const v2f*)(arow + k + koff);
        const v2f b = *(const v2f*)(brow + k + koff);
        acc = __builtin_amdgcn_wmma_f32_16x16x4_f32(
            /*neg_a=*/false, a, /*neg_b=*/false, b,
            /*c_mod=*/(short)0, acc, /*reuse_a=*/false, /*reuse_b=*/false);
    }

    // ---- post-process: clip, log-sigmoid, masks ----
    // D layout: lane L holds column N = L%16; VGPR r holds row M = r + 8*(L/16)
    const float cm = (rid < NC) ? cmask[rid] : 0.0f;   // fold ctx mask into col
    float sg = 0.0f, sc = 0.0f;
    if (rid >= NC) { sg = smask[rid - NC]; sc = scmask[rid - NC]; }
    const int mhi = (lane >> 4) << 3;    // 0 or 8

    float part = 0.0f;
#pragma unroll
    for (int r = 0; r < 8; ++r) {
        const int   M = r + mhi;
        const float v = acc[r];
        if (M == 0 && rid < NC) {              // target-context score
            float s = fminf(10.0f, fmaxf(-10.0f, v * cm));
            part += log1pf(__expf(-s));        // -log_sigmoid(s)
        } else if (M >= 1 && M <= NC && rid >= NC) {  // context-info score
            float s = fminf(10.0f, fmaxf(-10.0f, v)) * sg;
            part += log1pf(__expf(-s)) * sc;
        }
    }

    // ---- wave32 reduction, then per-block partial ----
#pragma unroll
    for (int off = 16; off > 0; off >>= 1)
        part += __shfl_xor(part, off, 32);

    __shared__ float wsum[WPB];
    if (lane == 0) wsum[wave] = part;
    __syncthreads();
    if (threadIdx.x == 0) {
        float s = 0.0f;
#pragma unroll
        for (int w = 0; w < WPB; ++w) s += wsum[w];
        partials[blockIdx.x] = s;
    }
}

// Deterministic final reduction of 2048 block partials -> scalar.
__global__ __launch_bounds__(256) void hg2vec_reduce_kernel(
    const float* __restrict__ partials, float* __restrict__ out)
{
    __shared__ float sh[256];
    float s = 0.0f;
    for (int i = threadIdx.x; i < NBLK; i += 256) s += partials[i];
    sh[threadIdx.x] = s;
    __syncthreads();
#pragma unroll
    for (int o = 128; o > 0; o >>= 1) {
        if (threadIdx.x < o) sh[threadIdx.x] += sh[threadIdx.x + o];
        __syncthreads();
    }
    if (threadIdx.x == 0) out[0] = sh[0];
}

extern "C" void kernel_launch(void* const* d_in, const int* in_sizes, int n_in,
                              void* d_out, int out_size, void* d_ws, size_t ws_size,
                              hipStream_t stream) {
    const int*   pos_u  = (const int*)  d_in[0];
    const int*   pos_v  = (const int*)  d_in[1];
    const int*   info_v = (const int*)  d_in[2];
    const float* W_in   = (const float*)d_in[3];
    const float* W_out  = (const float*)d_in[4];
    const float* cmask  = (const float*)d_in[5];
    const float* smask  = (const float*)d_in[6];
    const float* scmask = (const float*)d_in[7];
    float* partials = (float*)d_ws;          // NBLK floats (8 KB)
    float* out      = (float*)d_out;

    hg2vec_wmma_kernel<<<NBLK, 256, 0, stream>>>(
        pos_u, pos_v, info_v, W_in, W_out, cmask, smask, scmask, partials);
    hg2vec_reduce_kernel<<<1, 256, 0, stream>>>(partials, out);
}